// Decoder_34849364640399
// MI455X (gfx1250) — compile-verified
//
#include <hip/hip_runtime.h>
#include <hip/hip_bf16.h>
#include <math.h>

// MI455X / gfx1250, wave32. FP32 WMMA 16x16x4 path (matches fp32 reference
// precision over the 256-step recurrence; weights stay resident in 192MB L2).
// Split-K across waves in each block to shorten the per-step critical path.

typedef __attribute__((ext_vector_type(2))) float v2f;
typedef __attribute__((ext_vector_type(8))) float v8f;

#define HIDN 1024
#define LATN 128
#define INPN 512
#define NB   128     // batch
#define SEQL 256
#define BH   (NB * HIDN)       // one layer's h (or c) slice: 131072 floats
#define LBH  (2 * BH)          // both layers: 262144 floats (1 MB)

__device__ __forceinline__ float sigf(float x) { return 1.0f / (1.0f + expf(-x)); }

// ---------------------------------------------------------------------------
// Init: dh = latent @ W_fc^T + b_fc  (128 x 4096), scattered into h/c state
// with the torch reshape(L,B,2H) mapping: row b, col kcol ->
//   slice s = 2*b + (kcol>=2048), q = kcol & 2047; q<1024 -> h[s*H+q], else c.
// One-time cost (K=128), no split-K needed.
// ---------------------------------------------------------------------------
__global__ __launch_bounds__(128) void fc_init_kernel(
    const float* __restrict__ latent, const float* __restrict__ Wfc,
    const float* __restrict__ bfc, float* __restrict__ h, float* __restrict__ c)
{
  const int wv   = threadIdx.x >> 5;
  const int lane = threadIdx.x & 31;
  const int lh   = lane >> 4, ln = lane & 15;
  const int m0   = blockIdx.y * 16;
  const int n0   = (blockIdx.x * 4 + wv) * 16;

  v8f acc = {};
  const float* ap = latent + (size_t)(m0 + ln) * LATN + 2 * lh;
  const float* bp = Wfc    + (size_t)(n0 + ln) * LATN + 2 * lh;
#pragma unroll
  for (int k = 0; k < LATN; k += 4) {
    v2f a = *(const v2f*)(ap + k);
    v2f b = *(const v2f*)(bp + k);
    acc = __builtin_amdgcn_wmma_f32_16x16x4_f32(false, a, false, b,
                                                (short)0, acc, false, false);
  }
  const int   kcol = n0 + ln;
  const float bias = bfc[kcol];
  const int   hi   = kcol >> 11;     // 0: first (l,b2) slice, 1: second
  const int   q    = kcol & 2047;
#pragma unroll
  for (int v = 0; v < 8; ++v) {
    const int b = m0 + v + 8 * lh;
    const int s = 2 * b + hi;        // flat slice index l*128+b2 == s
    const float val = acc[v] + bias;
    if (q < HIDN) h[(size_t)s * HIDN + q]          = val;
    else          c[(size_t)s * HIDN + (q - HIDN)] = val;
  }
}

// ---------------------------------------------------------------------------
// out_t = sigmoid(fc_in @ W_map^T + b_map), fc_in gathered from h/c state.
// Row b of fc_in = [h[2b], c[2b], h[2b+1], c[2b+1]] (1024 each).
// Split-K: each of the 4 waves in a block handles one 1024-wide segment of
// the K=4096 reduction (segment == gather region), partials summed in LDS.
// Grid: 32 n-tiles x 8 m-tiles = 256 blocks, 1024 waves total.
// ---------------------------------------------------------------------------
__global__ __launch_bounds__(128) void map_kernel(
    const float* __restrict__ hcur, const float* __restrict__ ccur,
    const float* __restrict__ Wm, const float* __restrict__ bm,
    float* __restrict__ outt)
{
  __shared__ float psum[4][16][17];   // +1 pad: avoid 2-way LDS bank conflict
  const int wv   = threadIdx.x >> 5;  // segment 0..3
  const int lane = threadIdx.x & 31;
  const int lh   = lane >> 4, ln = lane & 15;
  const int m0   = blockIdx.y * 16;
  const int n0   = blockIdx.x * 16;

  const int b  = m0 + ln;
  const int s0 = 2 * b, s1 = 2 * b + 1;
  const float* aseg[4] = { hcur + (size_t)s0 * HIDN, ccur + (size_t)s0 * HIDN,
                           hcur + (size_t)s1 * HIDN, ccur + (size_t)s1 * HIDN };

  v8f acc = {};
  const float* ap = aseg[wv] + 2 * lh;
  const float* bp = Wm + (size_t)(n0 + ln) * 4096 + wv * 1024 + 2 * lh;
#pragma unroll 8
  for (int k = 0; k < 1024; k += 4) {
    v2f a  = *(const v2f*)(ap + k);
    v2f bq = *(const v2f*)(bp + k);
    acc = __builtin_amdgcn_wmma_f32_16x16x4_f32(false, a, false, bq,
                                                (short)0, acc, false, false);
  }
#pragma unroll
  for (int v = 0; v < 8; ++v)
    psum[wv][v + 8 * lh][ln] = acc[v];
  __syncthreads();

#pragma unroll
  for (int e = threadIdx.x; e < 256; e += 128) {
    const int m = e >> 4, jj = e & 15;
    const float val = psum[0][m][jj] + psum[1][m][jj] +
                      psum[2][m][jj] + psum[3][m][jj] + bm[n0 + jj];
    outt[(size_t)(m0 + m) * INPN + (n0 + jj)] = sigf(val);
  }
}

// ---------------------------------------------------------------------------
// One LSTM cell (one layer). 8 waves/block: wave w -> gate g = w&3 and
// K-half = w>>2 over the combined reduction [x (Kx) | h (1024)].
// Gate partials reduced through LDS; activations + state update at the end.
// Grid: 64 j-tiles x 8 m-tiles = 512 blocks, 4096 waves total.
// ---------------------------------------------------------------------------
__global__ __launch_bounds__(256) void lstm_cell_kernel(
    const float* __restrict__ x, int Kx,
    const float* __restrict__ Wih, const float* __restrict__ Whh,
    const float* __restrict__ bih, const float* __restrict__ bhh,
    const float* __restrict__ hcur, const float* __restrict__ ccur,
    float* __restrict__ hnew, float* __restrict__ cnew)
{
  __shared__ float gsh[8][16][17];    // +1 pad: avoid LDS bank conflicts
  const int w    = threadIdx.x >> 5;
  const int g    = w & 3;             // gate: 0=i 1=f 2=g 3=o
  const int half = w >> 2;            // K-half: 0 or 1
  const int lane = threadIdx.x & 31;
  const int lh   = lane >> 4, ln = lane & 15;
  const int m0   = blockIdx.y * 16;
  const int j0   = blockIdx.x * 16;
  const int n0   = g * HIDN + j0;

  // combined-K range for this wave: [lo, lo+KH) over [x | h]
  const int KH = (Kx + HIDN) >> 1;
  const int lo = half * KH, hi = lo + KH;
  const int xs = (lo < Kx) ? lo : Kx;
  const int xe = (hi < Kx) ? hi : Kx;
  const int hs = (lo > Kx) ? (lo - Kx) : 0;
  const int he = hi - Kx;             // > 0 when this half's range reaches h (both halves if Kx=512)

  v8f acc = {};
  {
    const float* ap = x   + (size_t)(m0 + ln) * Kx + 2 * lh;
    const float* bp = Wih + (size_t)(n0 + ln) * Kx + 2 * lh;
#pragma unroll 8
    for (int k = xs; k < xe; k += 4) {
      v2f a = *(const v2f*)(ap + k);
      v2f b = *(const v2f*)(bp + k);
      acc = __builtin_amdgcn_wmma_f32_16x16x4_f32(false, a, false, b,
                                                  (short)0, acc, false, false);
    }
  }
  if (he > 0) {
    const float* ap = hcur + (size_t)(m0 + ln) * HIDN + 2 * lh;
    const float* bp = Whh  + (size_t)(n0 + ln) * HIDN + 2 * lh;
#pragma unroll 8
    for (int k = hs; k < he; k += 4) {
      v2f a = *(const v2f*)(ap + k);
      v2f b = *(const v2f*)(bp + k);
      acc = __builtin_amdgcn_wmma_f32_16x16x4_f32(false, a, false, b,
                                                  (short)0, acc, false, false);
    }
  }
#pragma unroll
  for (int v = 0; v < 8; ++v)
    gsh[w][v + 8 * lh][ln] = acc[v];
  __syncthreads();

  // 256 threads, one gate-column element each
  {
    const int e  = threadIdx.x;
    const int m  = e >> 4, jj = e & 15;
    const int j  = j0 + jj;
    const size_t idx = (size_t)(m0 + m) * HIDN + j;
    const float gi = gsh[0][m][jj] + gsh[4][m][jj] + bih[0 * HIDN + j] + bhh[0 * HIDN + j];
    const float gf = gsh[1][m][jj] + gsh[5][m][jj] + bih[1 * HIDN + j] + bhh[1 * HIDN + j];
    const float gg = gsh[2][m][jj] + gsh[6][m][jj] + bih[2 * HIDN + j] + bhh[2 * HIDN + j];
    const float go = gsh[3][m][jj] + gsh[7][m][jj] + bih[3 * HIDN + j] + bhh[3 * HIDN + j];
    const float iv = sigf(gi);
    const float fv = sigf(gf);
    const float gv = tanhf(gg);
    const float ov = sigf(go);
    const float c2 = fv * ccur[idx] + iv * gv;
    cnew[idx] = c2;
    hnew[idx] = ov * tanhf(c2);
  }
}

// ---------------------------------------------------------------------------
extern "C" void kernel_launch(void* const* d_in, const int* in_sizes, int n_in,
                              void* d_out, int out_size, void* d_ws, size_t ws_size,
                              hipStream_t stream) {
  const float* latent = (const float*)d_in[0];
  // d_in[1] = seq_length (==256, fixed)
  const float* Wfc  = (const float*)d_in[2];
  const float* bfc  = (const float*)d_in[3];
  const float* Wmap = (const float*)d_in[4];
  const float* bmap = (const float*)d_in[5];
  const float* Wih0 = (const float*)d_in[6];
  const float* Whh0 = (const float*)d_in[7];
  const float* bih0 = (const float*)d_in[8];
  const float* bhh0 = (const float*)d_in[9];
  const float* Wih1 = (const float*)d_in[10];
  const float* Whh1 = (const float*)d_in[11];
  const float* bih1 = (const float*)d_in[12];
  const float* bhh1 = (const float*)d_in[13];
  float* out = (float*)d_out;

  // Ping-pong LSTM state in scratch: 4 x 1MB (h/c, two copies), layout [l][b][q].
  float* ws    = (float*)d_ws;
  float* h0buf = ws;
  float* c0buf = ws + LBH;
  float* h1buf = ws + 2 * LBH;
  float* c1buf = ws + 3 * LBH;

  fc_init_kernel<<<dim3(64, 8), 128, 0, stream>>>(latent, Wfc, bfc, h0buf, c0buf);

  for (int t = 0; t < SEQL; ++t) {
    float* hc = (t & 1) ? h1buf : h0buf;
    float* cc = (t & 1) ? c1buf : c0buf;
    float* hn = (t & 1) ? h0buf : h1buf;
    float* cn = (t & 1) ? c0buf : c1buf;
    float* ot = out + (size_t)t * NB * INPN;

    map_kernel<<<dim3(32, 8), 128, 0, stream>>>(hc, cc, Wmap, bmap, ot);

    // layer 0: x = out_t (K=512)
    lstm_cell_kernel<<<dim3(64, 8), 256, 0, stream>>>(
        ot, INPN, Wih0, Whh0, bih0, bhh0,
        hc, cc, hn, cn);

    // layer 1: x = new h0 (K=1024)
    lstm_cell_kernel<<<dim3(64, 8), 256, 0, stream>>>(
        hn, HIDN, Wih1, Whh1, bih1, bhh1,
        hc + BH, cc + BH, hn + BH, cn + BH);
  }
}